// GraphLayer_2783138808389
// MI455X (gfx1250) — compile-verified
//
#include <hip/hip_runtime.h>

// FiGNN-style fused graph layer for MI455X (gfx1250).
//   h_out = einsum('fde,bfe->bfd', W_out, h)       (per-field 64x64 GEMM)
//   aggr  = einsum('bfg,bgd->bfd', g, h_out)       (per-sample 40x40 x 40x64)
//   a     = einsum('fde,bfe->bfd', W_in, aggr) + bias
// All f32, memory-bound (~220MB @ 23.3TB/s ~ 9.5us); use V_WMMA_F32_16X16X4_F32.

#define NF 40
#define ED 64
#define TB 16          // batch samples per workgroup
#define NB_TOTAL 8192
#define THREADS 256    // 8 waves (wave32)

typedef __attribute__((ext_vector_type(2))) float v2f;
typedef __attribute__((ext_vector_type(4))) float v4f;
typedef __attribute__((ext_vector_type(8))) float v8f;
typedef int v4i __attribute__((vector_size(16)));   // matches builtin param type

__global__ __launch_bounds__(THREADS)
void fignn_fused(const float* __restrict__ g,
                 const float* __restrict__ h,
                 const float* __restrict__ Wout,
                 const float* __restrict__ Win,
                 const float* __restrict__ bias,
                 float* __restrict__ out)
{
    // LDS tile: [TB][NF][ED] floats = 160KB. Holds h, then h_out (in place),
    // then aggr (in place, per-sample exclusive overwrite).
    extern __shared__ float lds[];

    const int tid  = threadIdx.x;
    const int lane = tid & 31;
    const int wave = tid >> 5;
    const int b0   = blockIdx.x * TB;

    const int mrow  = lane & 15;         // M / N index within a 16-wide tile
    const int khalf = (lane >> 4) << 1;  // 0 for lanes 0-15, 2 for lanes 16-31
    const int mhi   = (lane >> 4) << 3;  // +8 row offset for C/D in upper lanes

    // ---------------- stage h tile into LDS (async global->LDS) -------------
    {
        const float* src = h + (size_t)b0 * NF * ED;
        const int n4 = TB * NF * ED / 4;   // 10240 16B chunks
#if __has_builtin(__builtin_amdgcn_global_load_async_to_lds_b128) && \
    __has_builtin(__builtin_amdgcn_s_wait_asynccnt)
        for (int i = tid; i < n4; i += THREADS) {
            __builtin_amdgcn_global_load_async_to_lds_b128(
                (__attribute__((address_space(1))) v4i*)(src + i * 4),
                (__attribute__((address_space(3))) v4i*)(&lds[i * 4]),
                0, 0);
        }
        __builtin_amdgcn_s_wait_asynccnt(0);
#else
        const v4f* s4 = (const v4f*)src;
        v4f*       d4 = (v4f*)lds;
        for (int i = tid; i < n4; i += THREADS) d4[i] = s4[i];
#endif
    }
    __syncthreads();

    // ---------------- phase 0: h_out[s][f][:] = W_out[f] @ h[s][f][:] -------
    // wave w owns fields [5w, 5w+5); in-place update of its field slices.
    for (int fi = 0; fi < 5; ++fi) {
        const int f = wave * 5 + fi;
        // A fragments: A[m=sample][k=e] from LDS, 16 k-steps of 4
        v2f afrag[16];
        #pragma unroll
        for (int kb = 0; kb < 16; ++kb)
            afrag[kb] = *(const v2f*)&lds[(mrow * NF + f) * ED + kb * 4 + khalf];

        v8f acc[4];
        #pragma unroll
        for (int nb = 0; nb < 4; ++nb) acc[nb] = (v8f){};

        #pragma unroll
        for (int nb = 0; nb < 4; ++nb) {
            #pragma unroll
            for (int kb = 0; kb < 16; ++kb) {
                // B[k][n] = W_out[f][n][k]: two consecutive e-values per lane
                const float* bp =
                    Wout + ((size_t)f * ED + nb * 16 + mrow) * ED + kb * 4 + khalf;
                v2f bfrag = *(const v2f*)bp;
                acc[nb] = __builtin_amdgcn_wmma_f32_16x16x4_f32(
                    false, afrag[kb], false, bfrag, (short)0, acc[nb], false, false);
            }
        }
        // store h_out tile back over the h slice (exclusive to this wave)
        #pragma unroll
        for (int nb = 0; nb < 4; ++nb) {
            #pragma unroll
            for (int r = 0; r < 8; ++r) {
                const int m = r + mhi;  // sample row
                lds[(m * NF + f) * ED + nb * 16 + mrow] = acc[nb][r];
            }
        }
    }
    __syncthreads();

    // ---------------- phase 1: aggr[s] = g[s] (40x40) @ h_out[s] (40x64) ----
    // wave w owns samples {2w, 2w+1}; overwrites its h_out slices with aggr.
    for (int si = 0; si < 2; ++si) {
        const int s = wave * 2 + si;
        const size_t gbase = (size_t)(b0 + s) * NF * NF;

        v8f acc[3][4];
        #pragma unroll
        for (int mb = 0; mb < 3; ++mb)
            #pragma unroll
            for (int nb = 0; nb < 4; ++nb) acc[mb][nb] = (v8f){};

        #pragma unroll
        for (int kb = 0; kb < NF; kb += 4) {      // 10 k-steps over g-fields
            v2f bfrag[4];
            #pragma unroll
            for (int nb = 0; nb < 4; ++nb) {      // B[k][n] = h_out[s][k][n]
                const int k0 = kb + khalf;
                bfrag[nb] = (v2f){ lds[(s * NF + k0    ) * ED + nb * 16 + mrow],
                                   lds[(s * NF + k0 + 1) * ED + nb * 16 + mrow] };
            }
            #pragma unroll
            for (int mb = 0; mb < 3; ++mb) {      // A[m=f][k=g] = g[s][f][g]
                const int frow = mb * 16 + mrow;
                v2f afrag = (v2f){};
                if (frow < NF)
                    afrag = *(const v2f*)(g + gbase + (size_t)frow * NF + kb + khalf);
                #pragma unroll
                for (int nb = 0; nb < 4; ++nb)
                    acc[mb][nb] = __builtin_amdgcn_wmma_f32_16x16x4_f32(
                        false, afrag, false, bfrag[nb], (short)0, acc[mb][nb],
                        false, false);
            }
        }
        // overwrite this sample's slice with aggr (all reads above are done)
        #pragma unroll
        for (int mb = 0; mb < 3; ++mb)
            #pragma unroll
            for (int nb = 0; nb < 4; ++nb)
                #pragma unroll
                for (int r = 0; r < 8; ++r) {
                    const int frow = mb * 16 + r + mhi;
                    if (frow < NF)
                        lds[(s * NF + frow) * ED + nb * 16 + mrow] = acc[mb][nb][r];
                }
    }
    __syncthreads();

    // ---------------- phase 2: out = W_in[f] @ aggr + bias ------------------
    for (int fi = 0; fi < 5; ++fi) {
        const int f = wave * 5 + fi;
        v2f afrag[16];
        #pragma unroll
        for (int kb = 0; kb < 16; ++kb)
            afrag[kb] = *(const v2f*)&lds[(mrow * NF + f) * ED + kb * 4 + khalf];

        #pragma unroll
        for (int nb = 0; nb < 4; ++nb) {
            const float bval = bias[nb * 16 + mrow];  // d is constant per lane
            v8f acc;
            #pragma unroll
            for (int r = 0; r < 8; ++r) acc[r] = bval;

            #pragma unroll
            for (int kb = 0; kb < 16; ++kb) {
                const float* bp =
                    Win + ((size_t)f * ED + nb * 16 + mrow) * ED + kb * 4 + khalf;
                v2f bfrag = *(const v2f*)bp;
                acc = __builtin_amdgcn_wmma_f32_16x16x4_f32(
                    false, afrag[kb], false, bfrag, (short)0, acc, false, false);
            }
            #pragma unroll
            for (int r = 0; r < 8; ++r) {
                const int m = r + mhi;
                out[((size_t)(b0 + m) * NF + f) * ED + nb * 16 + mrow] = acc[r];
            }
        }
    }
}

extern "C" void kernel_launch(void* const* d_in, const int* in_sizes, int n_in,
                              void* d_out, int out_size, void* d_ws, size_t ws_size,
                              hipStream_t stream) {
    // setup_inputs() order: g, h, W_in, W_out, bias_p
    const float* g    = (const float*)d_in[0];
    const float* h    = (const float*)d_in[1];
    const float* Win  = (const float*)d_in[2];
    const float* Wout = (const float*)d_in[3];
    const float* bias = (const float*)d_in[4];
    float* out = (float*)d_out;

    const int lds_bytes = TB * NF * ED * (int)sizeof(float);  // 163840
    (void)hipFuncSetAttribute((const void*)fignn_fused,
                              hipFuncAttributeMaxDynamicSharedMemorySize, lds_bytes);

    dim3 grid(NB_TOTAL / TB), block(THREADS);
    fignn_fused<<<grid, block, lds_bytes, stream>>>(g, h, Wout, Win, bias, out);
}